// SANGraphHead4Reg_26225070309387
// MI455X (gfx1250) — compile-verified
//
#include <hip/hip_runtime.h>
#include <hip/hip_bf16.h>

typedef __attribute__((ext_vector_type(2))) float v2f;
typedef __attribute__((ext_vector_type(8))) float v8f;
typedef __attribute__((ext_vector_type(4))) unsigned int v4u;
typedef __attribute__((ext_vector_type(8))) int v8i;
typedef __attribute__((ext_vector_type(4))) int v4i;

#define NN  262144
#define DD  1024
#define GG  1024
#define NAt 17
#define KS  36   // stacked K: [net(17) | dev(17) | pad(2)], multiple of 4
#define EPSF 1e-5f

// ---------------------------------------------------------------------------
// TDM helper: async 2D tile load (rows x rowElems fp32) from global to LDS.
// D# built per CDNA5 ISA 8.3/8.4: group0 = {flags, lds_addr, global_addr, type},
// group1 = {data_size, tensor dims/strides, tile dims}. 2D only -> groups 2/3
// zero. Issued once per calling wave; completion via TENSORcnt.
// This toolchain exposes the 6-arg builtin form.
// ---------------------------------------------------------------------------
__device__ __forceinline__ void tdm_load_tile_2d(const float* gptr, unsigned int ldsByteAddr,
                                                 unsigned int rows, unsigned int rowElems,
                                                 unsigned int tensorDim0, unsigned int strideElems)
{
    unsigned long long ga = (unsigned long long)(uintptr_t)gptr;
    v4u g0;
    g0.x = 1u;                                                        // count=1 (valid user D#)
    g0.y = ldsByteAddr;                                               // LDS dest (bytes)
    g0.z = (unsigned int)ga;                                          // global_addr[31:0]
    g0.w = ((unsigned int)(ga >> 32) & 0x01FFFFFFu) | 0x80000000u;    // addr[56:32] | type=2
    v8i g1;
    g1[0] = 0x20000;                                                  // data_size = 2 (4 bytes)
    g1[1] = (int)((tensorDim0 & 0xFFFFu) << 16);                      // tensor_dim0[15:0] @63:48
    g1[2] = (int)((tensorDim0 >> 16) | ((rows & 0xFFFFu) << 16));     // dim0 hi | tensor_dim1 lo
    g1[3] = (int)((rows >> 16) | (rowElems << 16));                   // dim1 hi | tile_dim0
    g1[4] = (int)(rows & 0xFFFFu);                                    // tile_dim1 | tile_dim2=0
    g1[5] = (int)strideElems;                                         // tensor_dim0_stride lo32
    g1[6] = 0;                                                        // stride0 hi | stride1 lo
    g1[7] = 0;                                                        // stride1 hi
    v4i z4 = {0, 0, 0, 0};
    v8i z8 = {0, 0, 0, 0, 0, 0, 0, 0};
    __builtin_amdgcn_tensor_load_to_lds(g0, g1, z4, z4, z8, 0);
}

// ---------------------------------------------------------------------------
// Kernel 1: fused emb-GEMM (WMMA fp32 16x16x4) + x-add + segment-sum pooling
// Tile: 16 nodes x 128 cols per block (8 waves, each wave one 16x16 WMMA tile)
// ---------------------------------------------------------------------------
__global__ __launch_bounds__(256) void pool_kernel(
    const float* __restrict__ x, const float* __restrict__ node_attr,
    const int* __restrict__ node_type, const int* __restrict__ batch,
    const float* __restrict__ net_W, const float* __restrict__ net_b,
    const float* __restrict__ dev_W, const float* __restrict__ dev_b,
    const float* __restrict__ pin_emb,
    float* __restrict__ pooled, float* __restrict__ cnt)
{
    __shared__ float As[16][KS];     // stacked, type-gated attr rows
    __shared__ float Bs[KS][128];    // stacked [net_W ; dev_W ; 0] column tile
    __shared__ int   typeS[16];
    __shared__ int   batchS[16];
    __shared__ int   pinS[16];

    const int tid      = threadIdx.x;
    const int lane     = tid & 31;
    const int wave     = tid >> 5;
    const int nodeBase = blockIdx.x * 16;
    const int colBase  = blockIdx.y * 128;

    // Wave 0: kick off TDM DMA of both weight tiles into Bs while the other
    // waves stage the attribute tile. rows 0..16 <- net_W, rows 17..33 <- dev_W.
    if (wave == 0) {
        unsigned int bsBase = (unsigned int)(uintptr_t)&Bs[0][0];
        tdm_load_tile_2d(net_W + colBase, bsBase,                 NAt, 128, DD, DD);
        tdm_load_tile_2d(dev_W + colBase, bsBase + NAt * 128 * 4, NAt, 128, DD, DD);
    }

    // Prefetch the x tile we will read in the epilogue (global_prefetch_b8)
    if (tid < 64) {
        int r = tid >> 2, s = tid & 3;
        __builtin_prefetch(&x[(size_t)(nodeBase + r) * DD + colBase + s * 32], 0, 0);
    }

    if (tid < 16) {
        int node = nodeBase + tid;
        typeS[tid]  = node_type[node];
        batchS[tid] = batch[node];
        pinS[tid]   = (int)node_attr[(size_t)node * NAt];  // trunc like astype(int32)
    }
    __syncthreads();

    // A: 16 x KS, gated by node type so one GEMM covers both weight matrices
    for (int idx = tid; idx < 16 * KS; idx += 256) {
        int m = idx / KS, k = idx % KS;
        int t = typeS[m];
        float v = 0.0f;
        if (k < NAt)            { if (t == 0) v = node_attr[(size_t)(nodeBase + m) * NAt + k]; }
        else if (k < 2 * NAt)   { if (t == 1) v = node_attr[(size_t)(nodeBase + m) * NAt + (k - NAt)]; }
        As[m][k] = v;
    }
    // zero the two K-padding rows of Bs (TDM filled rows 0..33)
    {
        int idx = tid;                 // exactly 256 elements, one per thread
        Bs[2 * NAt + (idx >> 7)][idx & 127] = 0.0f;
    }
    if (wave == 0)
        __builtin_amdgcn_s_wait_tensorcnt(0);   // wave0's DMAs landed in LDS
    __syncthreads();

    const int hi   = (lane >> 4) & 1;     // lanes 16-31 hold K+2 / rows M+8
    const int ml   = lane & 15;
    const int wcol = wave * 16 + ml;      // column within the 128-wide tile

    v8f c = {};
    #pragma unroll
    for (int ks = 0; ks < KS / 4; ++ks) {
        v2f a, b;
        int ka = ks * 4 + hi * 2;
        a.x = As[ml][ka];   a.y = As[ml][ka + 1];
        b.x = Bs[ka][wcol]; b.y = Bs[ka + 1][wcol];
        c = __builtin_amdgcn_wmma_f32_16x16x4_f32(false, a, false, b,
                                                  (short)0, c, false, false);
    }

    const int col = colBase + wcol;
    const float nb = net_b[col];
    const float db = dev_b[col];
    float vals[8];
    #pragma unroll
    for (int v = 0; v < 8; ++v) {
        int   m  = v + hi * 8;
        int   t  = typeS[m];
        float xv = x[(size_t)(nodeBase + m) * DD + col];
        float pv = pin_emb[(size_t)pinS[m] * DD + col];   // unconditional: branchless select
        float e  = (t == 0) ? c[v] + nb
                 : (t == 1) ? c[v] + db
                 : (t == 2) ? pv : 0.0f;
        vals[v] = xv + e;
    }

    // batch is sorted: usually the whole 16-node tile is one group
    if (batchS[0] == batchS[15]) {
        float ps = 0.0f;
        #pragma unroll
        for (int v = 0; v < 8; ++v) ps += vals[v];
        ps += __shfl_xor(ps, 16, 32);   // combine rows 0-7 with rows 8-15
        if (hi == 0)
            atomicAdd(&pooled[(size_t)batchS[0] * DD + col], ps);
    } else {
        #pragma unroll
        for (int v = 0; v < 8; ++v) {
            int m = v + hi * 8;
            atomicAdd(&pooled[(size_t)batchS[m] * DD + col], vals[v]);
        }
    }

    if (blockIdx.y == 0 && tid < 16)
        atomicAdd(&cnt[batchS[tid]], 1.0f);
}

// ---------------------------------------------------------------------------
// pooled[i] /= max(cnt, 1)
// ---------------------------------------------------------------------------
__global__ void finalize_pool_kernel(float* pooled, const float* cnt, int total)
{
    int i = blockIdx.x * 256 + threadIdx.x;
    if (i >= total) return;
    int g = i / DD;
    pooled[i] = pooled[i] / fmaxf(cnt[g], 1.0f);
}

// ---------------------------------------------------------------------------
// H = [relu](A @ W + b), WMMA fp32 16x16x4. Tile 16 rows x 128 cols per block.
// ---------------------------------------------------------------------------
__global__ __launch_bounds__(256) void gemm_bias_relu_kernel(
    const float* __restrict__ A, const float* __restrict__ W,
    const float* __restrict__ bias, float* __restrict__ H,
    int K, int Ncols, int doRelu)
{
    const int tid  = threadIdx.x;
    const int lane = tid & 31, wave = tid >> 5;
    const int hi   = (lane >> 4) & 1, ml = lane & 15;
    const int rm   = blockIdx.x * 16;
    const int col  = blockIdx.y * 128 + wave * 16 + ml;

    v8f c = {};
    for (int ks = 0; ks < (K >> 2); ++ks) {
        int k0 = ks * 4 + hi * 2;
        v2f a, b;
        a.x = A[(size_t)(rm + ml) * K + k0];
        a.y = A[(size_t)(rm + ml) * K + k0 + 1];
        b.x = W[(size_t)k0 * Ncols + col];
        b.y = W[(size_t)(k0 + 1) * Ncols + col];
        c = __builtin_amdgcn_wmma_f32_16x16x4_f32(false, a, false, b,
                                                  (short)0, c, false, false);
    }
    float bb = bias[col];
    #pragma unroll
    for (int v = 0; v < 8; ++v) {
        int m = v + hi * 8;
        float val = c[v] + bb;
        if (doRelu) val = fmaxf(val, 0.0f);
        H[(size_t)(rm + m) * Ncols + col] = val;
    }
}

// ---------------------------------------------------------------------------
// Per-column mean / rsqrt(var+eps) over M rows (biased var, matches jnp.var)
// ---------------------------------------------------------------------------
__global__ __launch_bounds__(256) void colstats_kernel(
    const float* __restrict__ H, float* __restrict__ meanv,
    float* __restrict__ rvarv, int M, int Ncols)
{
    __shared__ float s1[256], s2[256];
    int col = blockIdx.x;
    float a = 0.0f, b = 0.0f;
    for (int r = threadIdx.x; r < M; r += 256) {
        float v = H[(size_t)r * Ncols + col];
        a += v; b += v * v;
    }
    s1[threadIdx.x] = a; s2[threadIdx.x] = b;
    __syncthreads();
    for (int s = 128; s > 0; s >>= 1) {
        if (threadIdx.x < s) {
            s1[threadIdx.x] += s1[threadIdx.x + s];
            s2[threadIdx.x] += s2[threadIdx.x + s];
        }
        __syncthreads();
    }
    if (threadIdx.x == 0) {
        float m   = s1[0] / (float)M;
        float var = s2[0] / (float)M - m * m;
        meanv[col] = m;
        rvarv[col] = rsqrtf(var + EPSF);
    }
}

__global__ void bn_apply_kernel(float* H, const float* meanv, const float* rvarv,
                                const float* g, const float* b, int total, int Ncols)
{
    int i = blockIdx.x * 256 + threadIdx.x;
    if (i >= total) return;
    int col = i % Ncols;
    H[i] = (H[i] - meanv[col]) * rvarv[col] * g[col] + b[col];
}

// ---------------------------------------------------------------------------
// pred = h1 @ fc2_W + fc2_b ; out = [pred, y_reg]
// ---------------------------------------------------------------------------
__global__ void head_kernel(const float* __restrict__ H, const float* __restrict__ W,
                            const float* __restrict__ b, const float* __restrict__ y_reg,
                            float* __restrict__ out, int K, int Gn)
{
    int g = blockIdx.x * blockDim.x + threadIdx.x;
    if (g >= Gn) return;
    float acc = b[0];
    for (int k = 0; k < K; ++k) acc += H[(size_t)g * K + k] * W[k];
    out[g]      = acc;
    out[Gn + g] = y_reg[g];
}

__global__ void zero_kernel(float* p, int n)
{
    int i = blockIdx.x * 256 + threadIdx.x;
    if (i < n) p[i] = 0.0f;
}

// ---------------------------------------------------------------------------
extern "C" void kernel_launch(void* const* d_in, const int* in_sizes, int n_in,
                              void* d_out, int out_size, void* d_ws, size_t ws_size,
                              hipStream_t stream)
{
    const float* x         = (const float*)d_in[0];
    const float* node_attr = (const float*)d_in[1];
    const int*   node_type = (const int*)  d_in[2];
    const int*   batch     = (const int*)  d_in[3];
    const float* y_reg     = (const float*)d_in[4];
    const float* net_W     = (const float*)d_in[5];
    const float* net_b     = (const float*)d_in[6];
    const float* dev_W     = (const float*)d_in[7];
    const float* dev_b     = (const float*)d_in[8];
    const float* pin_emb   = (const float*)d_in[9];
    const float* fc0_W     = (const float*)d_in[10];
    const float* fc0_b     = (const float*)d_in[11];
    const float* fc1_W     = (const float*)d_in[12];
    const float* fc1_b     = (const float*)d_in[13];
    const float* fc2_W     = (const float*)d_in[14];
    const float* fc2_b     = (const float*)d_in[15];
    const float* bn0_g     = (const float*)d_in[16];
    const float* bn0_b     = (const float*)d_in[17];
    const float* bn1_g     = (const float*)d_in[18];
    const float* bn1_b     = (const float*)d_in[19];
    float* out = (float*)d_out;

    float* ws     = (float*)d_ws;
    float* pooled = ws;                               // G*D
    float* cnt    = pooled + (size_t)GG * DD;         // G
    float* h0     = cnt + GG;                         // G*512
    float* h1     = h0 + (size_t)GG * 512;            // G*256
    float* mean0  = h1 + (size_t)GG * 256;            // 512
    float* rvar0  = mean0 + 512;                      // 512
    float* mean1  = rvar0 + 512;                      // 256
    float* rvar1  = mean1 + 256;                      // 256

    // 1) zero accumulators (workspace is not re-poisoned between replays)
    int nz = GG * DD + GG;
    zero_kernel<<<(nz + 255) / 256, 256, 0, stream>>>(pooled, nz);

    // 2) fused embedding GEMM + x-add + segment-sum pool
    pool_kernel<<<dim3(NN / 16, DD / 128), 256, 0, stream>>>(
        x, node_attr, node_type, batch, net_W, net_b, dev_W, dev_b, pin_emb,
        pooled, cnt);

    // 3) mean-pool divide
    finalize_pool_kernel<<<(GG * DD + 255) / 256, 256, 0, stream>>>(pooled, cnt, GG * DD);

    // 4) fc0 + relu (1024x1024 @ 1024x512)
    gemm_bias_relu_kernel<<<dim3(GG / 16, 512 / 128), 256, 0, stream>>>(
        pooled, fc0_W, fc0_b, h0, DD, 512, 1);

    // 5) batchnorm 0
    colstats_kernel<<<512, 256, 0, stream>>>(h0, mean0, rvar0, GG, 512);
    bn_apply_kernel<<<(GG * 512 + 255) / 256, 256, 0, stream>>>(
        h0, mean0, rvar0, bn0_g, bn0_b, GG * 512, 512);

    // 6) fc1 + relu (1024x512 @ 512x256)
    gemm_bias_relu_kernel<<<dim3(GG / 16, 256 / 128), 256, 0, stream>>>(
        h0, fc1_W, fc1_b, h1, 512, 256, 1);

    // 7) batchnorm 1
    colstats_kernel<<<256, 256, 0, stream>>>(h1, mean1, rvar1, GG, 256);
    bn_apply_kernel<<<(GG * 256 + 255) / 256, 256, 0, stream>>>(
        h1, mean1, rvar1, bn1_g, bn1_b, GG * 256, 256);

    // 8) final projection + y_reg pass-through
    head_kernel<<<(GG + 255) / 256, 256, 0, stream>>>(h1, fc2_W, fc2_b, y_reg, out, 256, GG);
}